// ExponentialUnitNorm_35338990911851
// MI455X (gfx1250) — compile-verified
//
#include <hip/hip_runtime.h>

// EMA-normalization as a chunked scan-via-matmul on CDNA5 WMMA.
//
//   s_t = (1-a)|x_t| + a*s_{t-1},  out_t = x_t * rsqrt(s_t)
//
// Chunk 16 timesteps x 16 chains:  S = A*M + p (x) s_in
//   A[t][k] = (t>=k) ? (1-a)*a^(t-k) : 0   (16x16 lower triangular, constant)
//   p[t]    = a^(t+1)
// computed as 4 chained V_WMMA_F32_16X16X4_F32 with the p*s_in term in C.
// 62 full chunks in a clamp-free main loop; one 8-step epilogue chunk.

typedef __attribute__((ext_vector_type(2))) float v2f;
typedef __attribute__((ext_vector_type(8))) float v8f;

#define ALPHA_F 0.99f
#define OMA_F   0.01f
#define EPS_F   1e-14f
#define NB 16
#define NC 2
#define NT 1000
#define NF 481
#define NFT 31            // ceil(481/16)
#define NT_MAIN 992       // 62 full chunks of 16; epilogue handles t=992..999

__device__ __forceinline__ float pow_alpha(int e) {
    // alpha^e for 0 <= e <= 16, branch-free (runs once per wave at setup)
    float r = 1.0f;
#pragma unroll
    for (int i = 0; i < 16; ++i) r *= (i < e) ? ALPHA_F : 1.0f;
    return r;
}

__global__ void __launch_bounds__(256)
ema_unitnorm_wmma_kernel(const float2* __restrict__ x,
                         const float*  __restrict__ init_state,
                         float2*       __restrict__ out) {
    const int lane  = threadIdx.x & 31;
    const int wid   = threadIdx.x >> 5;
    const int ftile = blockIdx.x * 8 + wid;
    if (ftile >= NFT) return;              // uniform per wave: EXEC stays all-1s

    const int h  = lane >> 4;              // half-wave (0: lanes 0-15, 1: 16-31)
    const int j  = lane & 15;              // chain column N = lane%16
    const int m  = j;                      // A-matrix row M for this lane
    const int f0 = ftile * 16;
    const int f  = f0 + j;
    const bool fvalid = (f < NF);
    const int fc = fvalid ? f : (NF - 1);
    const size_t bcBase = (size_t)blockIdx.y * (size_t)NT * (size_t)NF;

    // --- constant A fragments: a[v] = A[m][4*(v>>1) + (v&1) + 2h] -------------
    float a_frag[8];
#pragma unroll
    for (int v = 0; v < 8; ++v) {
        const int k = 4 * (v >> 1) + (v & 1) + 2 * h;
        a_frag[v] = (m >= k) ? OMA_F * pow_alpha(m - k) : 0.0f;
    }
    // --- p factors for the C init: C row M = r + 8h gets alpha^(M+1) ---------
    float pfac[8];
#pragma unroll
    for (int r = 0; r < 8; ++r) pfac[r] = pow_alpha(r + 8 * h + 1);

    float s_in = init_state[fc];           // s0 broadcast per frequency bin

    // =================== main loop: 62 clamp-free full chunks ================
    for (int t0 = 0; t0 < NT_MAIN; t0 += 16) {
        // prefetch next 16x16 tile (2 cachelines per row via 32 lanes)
        {
            const int pr  = lane >> 1;
            const int pf0 = f0 + (lane & 1) * 16;
            const int pf  = (pf0 < NF) ? pf0 : (NF - 1);
            __builtin_prefetch(&x[bcBase + (size_t)(t0 + 16 + pr) * NF + pf], 0, 0);
        }

        // --- B fragments: magnitudes, b[v] = M[4*(v>>1)+(v&1)+2h][j] ---------
        float b_frag[8];
#pragma unroll
        for (int v = 0; v < 8; ++v) {
            const int k = 4 * (v >> 1) + (v & 1) + 2 * h;
            const float2 xv = x[bcBase + (size_t)(t0 + k) * NF + fc];
            const float d  = fmaxf(xv.x * xv.x + xv.y * xv.y, EPS_F);
            b_frag[v] = __builtin_amdgcn_sqrtf(d);   // raw v_sqrt_f32
        }

        // --- C init: p (outer) s_in ------------------------------------------
        v8f acc;
#pragma unroll
        for (int r = 0; r < 8; ++r) acc[r] = pfac[r] * s_in;

        // --- 4 chained f32 WMMAs: acc += A[:,4c:4c+4] * M[4c:4c+4,:] ---------
#pragma unroll
        for (int c4 = 0; c4 < 4; ++c4) {
            v2f av, bv;
            av[0] = a_frag[2 * c4];  av[1] = a_frag[2 * c4 + 1];
            bv[0] = b_frag[2 * c4];  bv[1] = b_frag[2 * c4 + 1];
            acc = __builtin_amdgcn_wmma_f32_16x16x4_f32(
                false, av, false, bv, (short)0, acc, false, false);
        }

        // --- outputs: out = x * rsqrt(S);  D row M = r + 8h, col = j ---------
#pragma unroll
        for (int r = 0; r < 8; ++r) {
            if (fvalid) {                       // loop-invariant lane mask
                const size_t idx = bcBase + (size_t)(t0 + r + 8 * h) * NF + f;
                const float2 xv = x[idx];
                const float inv = __builtin_amdgcn_rsqf(acc[r]);  // raw v_rsq_f32
                float2 o;
                o.x = xv.x * inv;
                o.y = xv.y * inv;
                out[idx] = o;
            }
        }

        // --- carry: s_out[j] = S[15][j] = acc[7] @ lane 16+j; broadcast ------
        const int sel = (((lane & 15) + 16) << 2);  // byte index for bpermute
        const int sv  = __builtin_amdgcn_ds_bpermute(sel, __float_as_int(acc[7]));
        s_in = __int_as_float(sv);
    }

    // =================== epilogue: last 8 timesteps (t = 992..999) ===========
    {
        const int t0 = NT_MAIN;
        const int tcnt = NT - NT_MAIN;      // 8

        float b_frag[8];
#pragma unroll
        for (int v = 0; v < 8; ++v) {
            const int k  = 4 * (v >> 1) + (v & 1) + 2 * h;
            const int kk = (k < tcnt) ? k : (tcnt - 1);
            const float2 xv = x[bcBase + (size_t)(t0 + kk) * NF + fc];
            const float d  = fmaxf(xv.x * xv.x + xv.y * xv.y, EPS_F);
            const float mg = __builtin_amdgcn_sqrtf(d);
            b_frag[v] = (k < tcnt) ? mg : 0.0f;
        }

        v8f acc;
#pragma unroll
        for (int r = 0; r < 8; ++r) acc[r] = pfac[r] * s_in;

#pragma unroll
        for (int c4 = 0; c4 < 4; ++c4) {
            v2f av, bv;
            av[0] = a_frag[2 * c4];  av[1] = a_frag[2 * c4 + 1];
            bv[0] = b_frag[2 * c4];  bv[1] = b_frag[2 * c4 + 1];
            acc = __builtin_amdgcn_wmma_f32_16x16x4_f32(
                false, av, false, bv, (short)0, acc, false, false);
        }

#pragma unroll
        for (int r = 0; r < 8; ++r) {
            const int trow = r + 8 * h;
            if (trow < tcnt && fvalid) {
                const size_t idx = bcBase + (size_t)(t0 + trow) * NF + f;
                const float2 xv = x[idx];
                const float inv = __builtin_amdgcn_rsqf(acc[r]);
                float2 o;
                o.x = xv.x * inv;
                o.y = xv.y * inv;
                out[idx] = o;
            }
        }
    }
}

extern "C" void kernel_launch(void* const* d_in, const int* in_sizes, int n_in,
                              void* d_out, int out_size, void* d_ws, size_t ws_size,
                              hipStream_t stream) {
    (void)in_sizes; (void)n_in; (void)out_size; (void)d_ws; (void)ws_size;
    const float2* x          = (const float2*)d_in[0];
    const float*  init_state = (const float*)d_in[1];
    float2*       out        = (float2*)d_out;

    // 31 f-tiles of 16 chains per (b,c); 8 waves per 256-thread block
    dim3 grid((NFT + 7) / 8, NB * NC);   // (4, 32)
    ema_unitnorm_wmma_kernel<<<grid, 256, 0, stream>>>(x, init_state, out);
}